// QGenModel_8108898255663
// MI455X (gfx1250) — compile-verified
//
#include <hip/hip_runtime.h>
#include <hip/hip_bf16.h>
#include <math.h>

// Problem constants (from reference): B=128, N=36, D=512, G=2
#define B_  128
#define N_  36
#define D_  512
#define G_  2
#define PAD 4
#define NTHREADS 256
#define NEGBIG (-1e10f)

typedef __attribute__((ext_vector_type(2))) float v2f;
typedef __attribute__((ext_vector_type(8))) float v8f;

// One workgroup per batch element b.
//   Stage 1: obj[b] (36x512 f32, 72KB) -> LDS (read once from HBM, reused 3x)
//   Stage 2: Wsum[g,d] = sum_m W[g,m,d];  P[g,d] = sum_m W[g,m,d]*obj[m,d]
//            packed as B'[g][k]: k<512 -> Wsum, k>=512 -> -P
//   Stage 3: logits[n,g] = sum_k A'[n][k]*B'[g][k], A'=[obj^2 | obj],
//            via V_WMMA_F32_16X16X4_F32 (3 M-tiles x 256 K-steps)
//   Stage 4: masked softmax over the 36 boxes (per g)
//   Stage 5: out[b,g,d] = sum_n w[n,g]*obj[n,d]
__launch_bounds__(NTHREADS)
__global__ void qgen_self_diff_attn(const float* __restrict__ obj,
                                    const float* __restrict__ W,
                                    const float* __restrict__ bias,
                                    const int*   __restrict__ mask,
                                    float*       __restrict__ out)
{
    __shared__ float s_obj[N_][D_ + PAD];     // 74.3 KB (stride 516: conflict-free A fetch)
    __shared__ float s_B[G_][2 * D_];         // 8 KB: [Wsum | -P]
    __shared__ float s_logits[N_ + 4][G_];
    __shared__ float s_w[N_][G_];

    const int b   = blockIdx.x;
    const int tid = threadIdx.x;

    // ---- Stage 1: stage obj[b] into LDS with coalesced b128 loads
    const float4* obj4 = (const float4*)(obj + (size_t)b * N_ * D_);
    for (int i = tid; i < (N_ * D_) / 4; i += NTHREADS) {
        float4 v = obj4[i];
        int idx = i * 4;
        int row = idx >> 9, col = idx & (D_ - 1);
        *(float4*)&s_obj[row][col] = v;       // row stride 2064B: 16B-aligned
    }
    __syncthreads();

    // ---- Stage 2: Wsum and P (W is 147KB -> L2-resident across all 128 blocks)
    for (int p = tid; p < G_ * D_; p += NTHREADS) {
        int g = p >> 9, d = p & (D_ - 1);
        const float* Wg = W + (size_t)g * (N_ * D_) + d;
        float ws = 0.f, pp = 0.f;
        #pragma unroll 4
        for (int m = 0; m < N_; ++m) {
            float w = Wg[m * D_];             // coalesced across d
            ws += w;
            pp += w * s_obj[m][d];            // conflict-free (consecutive d)
        }
        s_B[g][d]      = ws;
        s_B[g][D_ + d] = -pp;
    }
    __syncthreads();

    // ---- Stage 3: logits GEMM, M=36 (3 tiles), Npad=16 (2 used), K'=1024
    const int wave = tid >> 5;
    const int lane = tid & 31;
    if (wave < 3) {
        const int mt   = wave;
        const int ml   = lane & 15;                 // row within tile / col index
        const int mc   = min(mt * 16 + ml, N_ - 1); // clamp: junk rows never read
        const int nc   = min(ml, G_ - 1);           // clamp: junk cols never read
        const int koff = (lane >> 4) * 2;           // upper half-wave holds K+2,K+3
        v8f acc = {};
        // phase 0: A = obj^2 against Wsum
        for (int ks = 0; ks < D_ / 4; ++ks) {
            int kk = ks * 4 + koff;
            v2f a = *(const v2f*)&s_obj[mc][kk];
            a = a * a;
            v2f bb = *(const v2f*)&s_B[nc][kk];
            acc = __builtin_amdgcn_wmma_f32_16x16x4_f32(false, a, false, bb,
                                                        (short)0, acc, false, false);
        }
        // phase 1: A = obj against -P
        for (int ks = 0; ks < D_ / 4; ++ks) {
            int kk = ks * 4 + koff;
            v2f a  = *(const v2f*)&s_obj[mc][kk];
            v2f bb = *(const v2f*)&s_B[nc][D_ + kk];
            acc = __builtin_amdgcn_wmma_f32_16x16x4_f32(false, a, false, bb,
                                                        (short)0, acc, false, false);
        }
        // D layout: lane n = lane&15; rows m = v + 8*(lane>>4), v=0..7
        if (ml < G_) {
            int g  = ml;
            int hi = lane >> 4;
            float bg = bias[g];
            #pragma unroll
            for (int v = 0; v < 8; ++v) {
                int mg = mt * 16 + v + 8 * hi;
                if (mg < N_) s_logits[mg][g] = acc[v] + bg;
            }
        }
    }
    __syncthreads();

    // ---- Stage 4: masked softmax over box axis (36 elems per g; trivial)
    if (tid < G_) {
        int g = tid;
        const int* mrow = mask + b * N_;
        float vals[N_];
        float mx = -INFINITY;
        for (int m = 0; m < N_; ++m) {
            float l = mrow[m] ? s_logits[m][g] : NEGBIG;
            vals[m] = l;
            mx = fmaxf(mx, l);
        }
        float s = 0.f;
        for (int m = 0; m < N_; ++m) {
            float e = expf(vals[m] - mx);
            vals[m] = e;
            s += e;
        }
        float inv = 1.f / s;
        for (int m = 0; m < N_; ++m) s_w[m][g] = vals[m] * inv;
    }
    __syncthreads();

    // ---- Stage 5: vis_repr[b,g,d] = sum_n w[n,g]*obj[n,d]; coalesced store
    for (int p = tid; p < G_ * D_; p += NTHREADS) {
        int g = p >> 9, d = p & (D_ - 1);
        float acc = 0.f;
        #pragma unroll 4
        for (int m = 0; m < N_; ++m) acc += s_w[m][g] * s_obj[m][d];
        out[(size_t)b * (G_ * D_) + p] = acc;
    }
}

extern "C" void kernel_launch(void* const* d_in, const int* in_sizes, int n_in,
                              void* d_out, int out_size, void* d_ws, size_t ws_size,
                              hipStream_t stream) {
    (void)in_sizes; (void)n_in; (void)out_size; (void)d_ws; (void)ws_size;
    const float* obj  = (const float*)d_in[0];   // [B, N, D] f32
    const float* W    = (const float*)d_in[1];   // [G, N*D] f32
    const float* bias = (const float*)d_in[2];   // [G] f32
    const int*   mask = (const int*)d_in[3];     // [B, N] bool -> int
    float* out = (float*)d_out;                  // [B, G*D] f32

    qgen_self_diff_attn<<<B_, NTHREADS, 0, stream>>>(obj, W, bias, mask, out);
}